// CplAAD_19774029431254
// MI455X (gfx1250) — compile-verified
//
#include <hip/hip_runtime.h>
#include <cstdint>

#define TPB 256

// ---- CDNA5 async global->LDS staging (gfx1250) -----------------------------
// Each lane stages its own LDS slot; per-wave ASYNCcnt ordering suffices
// (no cross-lane LDS dependency), so s_wait_asynccnt alone synchronizes.
__device__ __forceinline__ void async_ld_b32(unsigned lds_addr, const float* gaddr) {
    asm volatile("global_load_async_to_lds_b32 %0, %1, off"
                 :: "v"(lds_addr), "v"(gaddr) : "memory");
}
__device__ __forceinline__ void wait_async_le2(void) {
    // up to two newer loads still in flight; the oldest two have completed
    asm volatile("s_wait_asynccnt 0x2" ::: "memory");
}
__device__ __forceinline__ void wait_async_0(void) {
    asm volatile("s_wait_asynccnt 0x0" ::: "memory");
}

struct PathState { float x, v, p1, p2, p3, p4, p5, p6, ir; };

__global__ __launch_bounds__(TPB)
void CplTS_mc_kernel(const float* __restrict__ x0,  const float* __restrict__ v0,
                     const float* __restrict__ f1,  const float* __restrict__ f2,
                     const float* __restrict__ f3,  const float* __restrict__ f4,
                     const float* __restrict__ f5,  const float* __restrict__ f6,
                     const float* __restrict__ kappa_p, const float* __restrict__ theta_p,
                     const float* __restrict__ rho_p,   const float* __restrict__ sigma_p,
                     const float* __restrict__ a0_p,    const float* __restrict__ a1_p,
                     const float* __restrict__ gamma_p, const float* __restrict__ varphi_p,
                     const float* __restrict__ strike_p,const float* __restrict__ delta_p,
                     const float* __restrict__ notional_p, const float* __restrict__ dt_p,
                     const float* __restrict__ Z,
                     float* __restrict__ out,
                     int nh, int steps)
{
    const int tid = threadIdx.x;
    const int i   = blockIdx.x * TPB + tid;
    if (i >= nh) return;

    // (zv, zp) interleaved per lane -> single ds_load_b64 readback,
    // lane L hits byte 8L: 32 lanes -> 32 distinct banks, conflict-free.
    __shared__ float2 zbufA[TPB];
    __shared__ float2 zbufB[TPB];

    // Raw LDS byte addresses (low 32 bits of the generic pointer == the
    // addrspace(3) offset the async instruction's VDST field expects).
    const unsigned lA0 = (unsigned)(uintptr_t)&zbufA[tid];
    const unsigned lA1 = lA0 + 4u;
    const unsigned lB0 = (unsigned)(uintptr_t)&zbufB[tid];
    const unsigned lB1 = lB0 + 4u;

    // Z is [steps][2][nh]; both channels stride by 2*nh per step.
    const float* gzv = Z + (size_t)i;            // Z[0][0][i]
    const float* gzp = Z + (size_t)nh + i;       // Z[0][1][i]
    const size_t stepStride = 2 * (size_t)nh;

    // Prologue: stage step 0 -> A, step 1 -> B.
    async_ld_b32(lA0, gzv);
    async_ld_b32(lA1, gzp);
    gzv += stepStride; gzp += stepStride;
    if (steps > 1) {
        async_ld_b32(lB0, gzv);
        async_ld_b32(lB1, gzp);
        gzv += stepStride; gzp += stepStride;
    }

    // ---- scalar parameters (uniform s_loads) ----
    const float kappa    = *kappa_p;
    const float theta    = *theta_p;
    const float rho      = *rho_p;
    const float sigma    = *sigma_p;
    const float a0       = *a0_p;
    const float a1       = *a1_p;
    const float g        = *gamma_p;
    const float varphi   = *varphi_p;
    const float strike   = *strike_p;
    const float delta    = *delta_p;
    const float notional = *notional_p;
    const float dt       = *dt_p;

    // ---- derived constants ----
    const float A     = a0 / g + a1 / (g * g);
    const float Bc    = a1 / g;
    const float sq1m  = __builtin_amdgcn_sqrtf(1.0f - rho * rho);
    const float sqdt  = __builtin_amdgcn_sqrtf(dt);
    const float c5    = a0 * Bc + a1 * A;
    const float c6    = a1 * Bc;
    const float Aa0   = A * a0;
    const float Aa1   = A * a1;
    const float sqdtR = sqdt * rho;
    const float sqdtS = sqdt * sq1m;
    const float omgdt  = 1.0f - g * dt;          // (1 - g dt)
    const float om2gdt = 1.0f - 2.0f * g * dt;   // (1 - 2g dt)
    const float omkdt  = 1.0f - kappa * dt;      // (1 - kappa dt)
    const float kthdt  = kappa * theta * dt;

    // ---- initial state (shared by antithetic pair) ----
    PathState sp, sm;
    sp.x = x0[i]; sp.v = v0[i];
    sp.p1 = f1[i]; sp.p2 = f2[i]; sp.p3 = f3[i];
    sp.p4 = f4[i]; sp.p5 = f5[i]; sp.p6 = f6[i];
    sp.ir = 0.0f;
    sm = sp;

    auto do_step = [&](PathState& S, float dWx, float dWv) {
        const float sv = __builtin_amdgcn_sqrtf(fmaxf(S.v, 0.0f));  // full truncation
        const float r  = varphi + a0 * S.x + a1 * S.p1
                       + Aa0 * (S.p2 - S.p3) + Aa1 * S.p4
                       - c5 * S.p5 - c6 * S.p6;
        S.ir = fmaf(r, dt, S.ir);
        const float xn = fmaf(S.x, omgdt, sv * dWx);
        const float vn = fmaf(S.v, omkdt, fmaf(sigma * sv, dWv, kthdt));
        const float q1 = fmaf(S.p1, omgdt,  S.x  * dt);
        const float q2 = fmaf(S.p2, omgdt,  S.v  * dt);
        const float q3 = fmaf(S.p3, om2gdt, S.v  * dt);
        const float q4 = fmaf(S.p4, omgdt,  S.p2 * dt);
        const float q5 = fmaf(S.p5, om2gdt, S.p3 * dt);
        const float q6 = fmaf(S.p6, om2gdt, (S.p5 + S.p5) * dt);
        S.x = xn; S.v = vn;
        S.p1 = q1; S.p2 = q2; S.p3 = q3; S.p4 = q4; S.p5 = q5; S.p6 = q6;
    };

    auto do_pair = [&](float zv, float zp) {
        const float dWv = sqdt * zv;
        const float dWx = fmaf(sqdtR, zv, sqdtS * zp);
        do_step(sp,  dWx,  dWv);      // + path
        do_step(sm, -dWx, -dWv);      // antithetic path (uses -Z)
    };

    // ---- main time loop: unrolled x2, statically double-buffered ----
    int s = 0;
    for (; s + 1 < steps; s += 2) {
        // step s (buffer A): A's two loads are the oldest outstanding
        wait_async_le2();
        const float2 zA = zbufA[tid];
        if (s + 2 < steps) {                  // refill A with step s+2
            async_ld_b32(lA0, gzv);
            async_ld_b32(lA1, gzp);
            gzv += stepStride; gzp += stepStride;
        }
        do_pair(zA.x, zA.y);

        // step s+1 (buffer B)
        if (s + 3 < steps) {
            wait_async_le2();                 // B's loads are the oldest two
            const float2 zB = zbufB[tid];
            async_ld_b32(lB0, gzv);           // refill B with step s+3
            async_ld_b32(lB1, gzp);
            gzv += stepStride; gzp += stepStride;
            do_pair(zB.x, zB.y);
        } else {
            wait_async_0();                   // B, and A's refill if issued, may be in flight
            const float2 zB = zbufB[tid];
            do_pair(zB.x, zB.y);
        }
    }
    if (s < steps) {                          // odd step count: last step in A
        wait_async_0();
        const float2 zA = zbufA[tid];
        do_pair(zA.x, zA.y);
    }

    // ---- terminal: ZCB bond loadings + caplet payoff ----
    const float tau = delta;
    const float e1  = __expf(-g * tau);
    const float e2  = __expf(-2.0f * g * tau);
    const float Bx  = -A + e1 * (A + Bc * tau);
    const float B1  = Bc * (e1 - 1.0f);
    const float B2  = A * Bx;
    const float B4  = A * B1;
    const float inv2g  = 1.0f / (2.0f * g);
    const float inv4g2 = 1.0f / (4.0f * g * g);
    const float inv4g3 = 1.0f / (4.0f * g * g * g);
    const float I0  = (1.0f - e2) * inv2g;
    const float I1  = (1.0f - e2 * (1.0f + 2.0f * g * tau)) * inv4g2;
    const float I2  = inv4g3 - e2 * (tau * tau * inv2g + tau * (2.0f * inv4g2) + inv4g3);
    const float B3  = a0 * A * I0 + (a0 * Bc + a1 * A) * I1 + a1 * Bc * I2;
    const float B5  = (a0 * Bc + a1 * A) * I0 + 2.0f * a1 * Bc * I1;
    const float B6  = a1 * Bc * I0;
    const float Kt  = 1.0f / (1.0f + delta * strike);
    const float fac = notional * (1.0f + delta * strike);

    auto payoff = [&](const PathState& S) -> float {
        const float logP = -varphi * tau + Bx * S.x + B1 * S.p1 + B2 * S.p2
                         + B3 * S.p3 + B4 * S.p4 + B5 * S.p5 + B6 * S.p6;
        const float pT = __expf(logP);
        const float pay = fac * fmaxf(Kt - pT, 0.0f);
        return pay * __expf(-S.ir);
    };

    out[i]      = payoff(sp);
    out[i + nh] = payoff(sm);
}

extern "C" void kernel_launch(void* const* d_in, const int* in_sizes, int n_in,
                              void* d_out, int out_size, void* d_ws, size_t ws_size,
                              hipStream_t stream) {
    const float* x0  = (const float*)d_in[0];
    const float* v0  = (const float*)d_in[1];
    const float* f1  = (const float*)d_in[2];
    const float* f2  = (const float*)d_in[3];
    const float* f3  = (const float*)d_in[4];
    const float* f4  = (const float*)d_in[5];
    const float* f5  = (const float*)d_in[6];
    const float* f6  = (const float*)d_in[7];
    const float* kap = (const float*)d_in[8];
    const float* the = (const float*)d_in[9];
    const float* rho = (const float*)d_in[10];
    const float* sig = (const float*)d_in[11];
    const float* a0  = (const float*)d_in[12];
    const float* a1  = (const float*)d_in[13];
    const float* gam = (const float*)d_in[14];
    const float* vph = (const float*)d_in[15];
    const float* stk = (const float*)d_in[16];
    const float* del = (const float*)d_in[17];
    const float* ntl = (const float*)d_in[18];
    const float* dt  = (const float*)d_in[19];
    const float* Z   = (const float*)d_in[20];
    float* out = (float*)d_out;

    const int nh    = in_sizes[0];
    const int steps = in_sizes[20] / (2 * nh);
    const int blocks = (nh + TPB - 1) / TPB;

    CplTS_mc_kernel<<<blocks, TPB, 0, stream>>>(
        x0, v0, f1, f2, f3, f4, f5, f6,
        kap, the, rho, sig, a0, a1, gam, vph,
        stk, del, ntl, dt, Z, out, nh, steps);
}